// CapsuleLayer_19447611916455
// MI455X (gfx1250) — compile-verified
//
#include <hip/hip_runtime.h>
#include <hip/hip_bf16.h>
#include <math.h>

// Problem constants (fixed by reference setup_inputs)
#define BATCH   64
#define N_IN    2048
#define N_OUT   32
#define D_IN    8
#define D_OUT   16
#define CI      8          // i-chunk per workgroup -> 2048/8 = 256 workgroups
#define EPSV    1e-7f

typedef __attribute__((ext_vector_type(2))) float v2f;
typedef __attribute__((ext_vector_type(8))) float v8f;

__device__ __forceinline__ float wave_max32(float x) {
    #pragma unroll
    for (int m = 16; m >= 1; m >>= 1) x = fmaxf(x, __shfl_xor(x, m, 32));
    return x;
}
__device__ __forceinline__ float wave_sum32(float x) {
    #pragma unroll
    for (int m = 16; m >= 1; m >>= 1) x += __shfl_xor(x, m, 32);
    return x;
}

// u[b, i, d] -> uT[i, d, b]   (coalesced writes; reads are gather but only 4MB total)
__global__ void transpose_u_kernel(const float* __restrict__ u, float* __restrict__ uT) {
    int t = blockIdx.x * blockDim.x + threadIdx.x;   // 0 .. 64*2048*8-1
    int b    = t & 63;
    int rest = t >> 6;
    int d    = rest & 7;
    int i    = rest >> 3;
    uT[t] = u[(b * N_IN + i) * D_IN + d];
}

// One routing iteration, fused:
//   for each i: u_hat tile via WMMA f32 16x16x4 (D[e x b] = W[i,j]^T x u^T),
//   logits b_ij = sum_{t<nv} u_hat . v_t   (recomputed, never stored),
//   softmax over j (cross-wave via LDS), s[b,j,e] += c * u_hat  (atomics at end).
__global__ __launch_bounds__(1024) void routing_pass_kernel(
    const float* __restrict__ uT,   // [N_IN][D_IN][BATCH]
    const float* __restrict__ Wm,   // [N_IN][N_OUT][D_IN][D_OUT]
    const float* __restrict__ v0,   // [BATCH][N_OUT][D_OUT] or unused
    const float* __restrict__ v1,   // [BATCH][N_OUT][D_OUT] or unused
    float* __restrict__ s,          // [BATCH][N_OUT][D_OUT], pre-zeroed
    int nv)                         // number of previous v's (0,1,2)
{
    __shared__ float lds_c[64 * 33];          // logits -> coupling coeffs (padded rows)

    const int tid  = threadIdx.x;
    const int w    = tid >> 5;                // wave 0..31
    const int lane = tid & 31;
    const int bt   = w & 3;                   // b-tile (16 batches)
    const int jg   = w >> 2;                  // j-group of 4 (0..7)
    const int hf   = lane >> 4;               // lane half: K/M split
    const int bl   = lane & 15;
    const int b    = bt * 16 + bl;            // this lane's batch column (B/C/D N-dim)
    const int el   = bl;                      // this lane's e row for A operand (M-dim)

    float sacc[4][8];
    #pragma unroll
    for (int jj = 0; jj < 4; ++jj)
        #pragma unroll
        for (int r = 0; r < 8; ++r) sacc[jj][r] = 0.f;

    const int i0 = blockIdx.x * CI;

    for (int ic = 0; ic < CI; ++ic) {
        const int i = i0 + ic;

        // B operand: B[d, b] = uT[i][d][b].  VGPR v, half hf -> K = v + 2*hf (+k0)
        const int ub = (i * D_IN) * BATCH + b;
        v2f blo, bhi;
        blo.x = uT[ub + (0 + 2 * hf) * BATCH];
        blo.y = uT[ub + (1 + 2 * hf) * BATCH];
        bhi.x = uT[ub + (4 + 2 * hf) * BATCH];
        bhi.y = uT[ub + (5 + 2 * hf) * BATCH];

        v8f   dt[4];
        float logit[4];

        #pragma unroll
        for (int jj = 0; jj < 4; ++jj) {
            const int j = jg * 4 + jj;
            // A operand: A[e, d] = W[i, j, d, e].
            const int wb = ((i * N_OUT + j) * D_IN) * D_OUT + el;
            v2f alo, ahi;
            alo.x = Wm[wb + (0 + 2 * hf) * D_OUT];
            alo.y = Wm[wb + (1 + 2 * hf) * D_OUT];
            ahi.x = Wm[wb + (4 + 2 * hf) * D_OUT];
            ahi.y = Wm[wb + (5 + 2 * hf) * D_OUT];

            v8f acc = {0.f, 0.f, 0.f, 0.f, 0.f, 0.f, 0.f, 0.f};
            acc = __builtin_amdgcn_wmma_f32_16x16x4_f32(false, alo, false, blo,
                                                        (short)0, acc, false, false);
            acc = __builtin_amdgcn_wmma_f32_16x16x4_f32(false, ahi, false, bhi,
                                                        (short)0, acc, false, false);
            dt[jj] = acc;

            // logit = sum over stored v's of  u_hat[b,j,:] . v[b,j,:]
            float lg = 0.f;
            for (int t = 0; t < nv; ++t) {
                const float* vt = (t == 0) ? v0 : v1;
                const float* vp = vt + ((b * N_OUT + j) * D_OUT + 8 * hf);
                float p = 0.f;
                #pragma unroll
                for (int r = 0; r < 8; ++r) p += acc[r] * vp[r];  // e = r + 8*hf
                lg += p;
            }
            lg += __shfl_xor(lg, 16, 32);      // combine the two e-halves
            logit[jj] = lg;
        }

        if (hf == 0) {
            #pragma unroll
            for (int jj = 0; jj < 4; ++jj) lds_c[b * 33 + jg * 4 + jj] = logit[jj];
        }
        __syncthreads();

        // softmax over j (32 entries/row); wave w handles rows 2w, 2w+1; lane = j
        #pragma unroll
        for (int rep = 0; rep < 2; ++rep) {
            const int row = w * 2 + rep;
            float x  = lds_c[row * 33 + lane];
            float mx = wave_max32(x);
            float ex = __expf(x - mx);
            float sm = wave_sum32(ex);
            lds_c[row * 33 + lane] = ex / sm;
        }
        __syncthreads();

        // s[b, j, e] += c[b, j] * u_hat[b, j, e]
        #pragma unroll
        for (int jj = 0; jj < 4; ++jj) {
            const float c = lds_c[b * 33 + jg * 4 + jj];
            #pragma unroll
            for (int r = 0; r < 8; ++r) sacc[jj][r] += c * dt[jj][r];
        }
        __syncthreads();   // lds_c reused next i
    }

    // flush local s to global (each (bt,jg) region unique per workgroup; atomics
    // across the 256 workgroups partitioning i)
    #pragma unroll
    for (int jj = 0; jj < 4; ++jj) {
        const int j = jg * 4 + jj;
        #pragma unroll
        for (int r = 0; r < 8; ++r) {
            const int e = r + 8 * hf;
            __hip_atomic_fetch_add(&s[(b * N_OUT + j) * D_OUT + e], sacc[jj][r],
                                   __ATOMIC_RELAXED, __HIP_MEMORY_SCOPE_AGENT);
        }
    }
}

// v = squash(s): one thread per (b, j)
__global__ void squash_kernel(const float* __restrict__ s, float* __restrict__ v) {
    const int p = blockIdx.x * blockDim.x + threadIdx.x;   // 0 .. 2047
    const float* sp = s + p * D_OUT;
    float tmp[D_OUT];
    float n2 = 0.f;
    #pragma unroll
    for (int e = 0; e < D_OUT; ++e) { tmp[e] = sp[e]; n2 += tmp[e] * tmp[e]; }
    const float scale = (n2 / (1.f + n2)) * rsqrtf(n2 + EPSV);
    #pragma unroll
    for (int e = 0; e < D_OUT; ++e) v[p * D_OUT + e] = tmp[e] * scale;
}

extern "C" void kernel_launch(void* const* d_in, const int* in_sizes, int n_in,
                              void* d_out, int out_size, void* d_ws, size_t ws_size,
                              hipStream_t stream) {
    const float* u  = (const float*)d_in[0];   // [64, 2048, 8]
    const float* Wm = (const float*)d_in[1];   // [2048, 32, 8, 16]
    float* out = (float*)d_out;                // [64, 32, 16]

    // workspace layout (floats):  uT | s | v0 | v1   (~4.5 MB total)
    float* uT = (float*)d_ws;
    float* sb = uT + (size_t)BATCH * N_IN * D_IN;            // 1048576
    float* v0 = sb + (size_t)BATCH * N_OUT * D_OUT;          // +32768
    float* v1 = v0 + (size_t)BATCH * N_OUT * D_OUT;          // +32768

    const int SVE = BATCH * N_OUT * D_OUT;                   // 32768

    transpose_u_kernel<<<(BATCH * N_IN * D_IN) / 256, 256, 0, stream>>>(u, uT);

    for (int pass = 0; pass < 3; ++pass) {
        hipMemsetAsync(sb, 0, (size_t)SVE * sizeof(float), stream);
        routing_pass_kernel<<<N_IN / CI, 1024, 0, stream>>>(uT, Wm, v0, v1, sb, pass);
        float* vdst = (pass == 0) ? v0 : (pass == 1) ? v1 : out;
        squash_kernel<<<(BATCH * N_OUT) / 256, 256, 0, stream>>>(sb, vdst);
    }
}